// GCN_68951404970424
// MI455X (gfx1250) — compile-verified
//
#include <hip/hip_runtime.h>
#include <hip/hip_bf16.h>

typedef float v2f __attribute__((ext_vector_type(2)));
typedef float v8f __attribute__((ext_vector_type(8)));

#define IN_F 128
#define HID 3

// ---------------------------------------------------------------------------
// Kernel 1: xw = x @ W via V_WMMA_F32_16X16X4_F32, one 16-row tile per wave.
// Also initializes deg[i] = 1.0f (accounts for the self-loop).
//
// B (W zero-padded to 16 cols) is staged in LDS as ldsB[pair][col][2]:
//   ldsB[p][col][j] = Wpad[2p + j][col],  p = K/2, j = 0..1
// so the per-step B fragment for lane (half h, lr) is one contiguous 8-byte
// read -> a single unconditional ds_load_b64 (no EXEC masking in the loop).
// Per ISA A-layout (16x4 f32): lane lr holds rows M=lr; half selects K pair,
// so the A fragment is a contiguous float2 -> one global_load_b64.
// ---------------------------------------------------------------------------
__global__ __launch_bounds__(256) void gcn_xw_wmma(
    const float* __restrict__ x, const float* __restrict__ W,
    float* __restrict__ xw, float* __restrict__ deg, int n_nodes) {
  __shared__ float ldsB[64 * 32];  // [pair 0..63][col 0..15][j 0..1], 8 KB
  const int t = threadIdx.x;
  for (int i = t; i < 64 * 32; i += 256) {
    const int rem = i & 31;
    const int col = rem >> 1;
    const int k   = ((i >> 5) << 1) | (rem & 1);  // K index = 2*pair + j
    ldsB[i] = (col < HID) ? W[k * HID + col] : 0.0f;
  }
  __syncthreads();

  const int gtid = blockIdx.x * 256 + t;
  if (gtid < n_nodes) deg[gtid] = 1.0f;  // self-loop contribution

  const int wave = gtid >> 5;   // global wave id == tile id
  const int lane = t & 31;
  const int half = lane >> 4;   // 0: lanes 0-15, 1: lanes 16-31
  const int lr   = lane & 15;
  const int row0 = wave * 16;
  if (row0 >= n_nodes) return;  // wave-uniform: EXEC stays all-1s for WMMA

  const float* xrow = x + (size_t)(row0 + lr) * IN_F + half * 2;
  const float* brow = ldsB + half * 32 + lr * 2;

  v8f acc = {};
#pragma unroll
  for (int k = 0; k < IN_F; k += 4) {
    v2f a = *(const v2f*)(xrow + k);              // A[M=lr][K=k+2h .. +1]
    v2f b = *(const v2f*)(brow + (k >> 1) * 32);  // B[K=k+2h .. +1][N=lr]
    acc = __builtin_amdgcn_wmma_f32_16x16x4_f32(
        /*neg_a=*/false, a, /*neg_b=*/false, b,
        /*c_mod=*/(short)0, acc, /*reuse_a=*/false, /*reuse_b=*/false);
  }

  // C/D layout: VGPR v -> row M = v + 8*half, column N = lr. Keep cols 0..3.
  if (lr < 4) {
#pragma unroll
    for (int v = 0; v < 8; ++v) {
      const int row = row0 + v + half * 8;
      xw[(size_t)row * 4 + lr] = acc[v];
    }
  }
}

// ---------------------------------------------------------------------------
// Kernel 2: deg[col[e]] += 1 over the 3.2M real edges (f32 HW atomics in L2).
// ---------------------------------------------------------------------------
__global__ __launch_bounds__(256) void gcn_degree(
    const long long* __restrict__ ei, float* __restrict__ deg, int n_edges) {
  const int e = blockIdx.x * blockDim.x + threadIdx.x;
  if (e < n_edges) {
    const int c = (int)ei[(size_t)n_edges + e];
    unsafeAtomicAdd(&deg[c], 1.0f);
  }
}

// ---------------------------------------------------------------------------
// Kernel 3: dinv = rsqrt(deg); seed agg with self-loop message xw[i]/deg[i].
// (deg >= 1 always because of the self-loop, so no zero-guard needed.)
// ---------------------------------------------------------------------------
__global__ __launch_bounds__(256) void gcn_selfloop(
    const float* __restrict__ deg, float* __restrict__ dinv,
    const float* __restrict__ xw, float* __restrict__ agg, int n_nodes) {
  const int i = blockIdx.x * blockDim.x + threadIdx.x;
  if (i < n_nodes) {
    const float d  = deg[i];
    const float di = rsqrtf(d);
    dinv[i] = di;
    const float inv = di * di;  // 1/deg
    float4 v = *(const float4*)(xw + (size_t)i * 4);
    v.x *= inv; v.y *= inv; v.z *= inv; v.w = 0.0f;
    *(float4*)(agg + (size_t)i * 4) = v;
  }
}

// ---------------------------------------------------------------------------
// Kernel 4: edge scatter: agg[c] += xw[r] * dinv[r] * dinv[c].
// float4 gather of xw (stride-4, 16B aligned) + 3 f32 atomics per edge.
// ---------------------------------------------------------------------------
__global__ __launch_bounds__(256) void gcn_scatter(
    const long long* __restrict__ ei, const float* __restrict__ dinv,
    const float* __restrict__ xw, float* __restrict__ agg, int n_edges) {
  const int e = blockIdx.x * blockDim.x + threadIdx.x;
  if (e < n_edges) {
    const int r = (int)ei[e];
    const int c = (int)ei[(size_t)n_edges + e];
    const float nrm = dinv[r] * dinv[c];
    const float4 v = *(const float4*)(xw + (size_t)r * 4);
    float* a = agg + (size_t)c * 4;
    unsafeAtomicAdd(a + 0, v.x * nrm);
    unsafeAtomicAdd(a + 1, v.y * nrm);
    unsafeAtomicAdd(a + 2, v.z * nrm);
  }
}

// ---------------------------------------------------------------------------
// Kernel 5: h = relu(agg + b); z = h @ Wout + bout.
// d_out = h (n*3 floats) followed by z (n*4 floats).
// ---------------------------------------------------------------------------
__global__ __launch_bounds__(256) void gcn_out(
    const float* __restrict__ agg, const float* __restrict__ b,
    const float* __restrict__ Wout, const float* __restrict__ bout,
    float* __restrict__ out, int n_nodes) {
  const int i = blockIdx.x * blockDim.x + threadIdx.x;
  if (i < n_nodes) {
    const float4 a = *(const float4*)(agg + (size_t)i * 4);
    const float h0 = fmaxf(a.x + b[0], 0.0f);
    const float h1 = fmaxf(a.y + b[1], 0.0f);
    const float h2 = fmaxf(a.z + b[2], 0.0f);
    out[(size_t)i * 3 + 0] = h0;
    out[(size_t)i * 3 + 1] = h1;
    out[(size_t)i * 3 + 2] = h2;
    float* z = out + (size_t)n_nodes * 3 + (size_t)i * 4;
#pragma unroll
    for (int k = 0; k < 4; ++k)
      z[k] = bout[k] + h0 * Wout[0 * 4 + k] + h1 * Wout[1 * 4 + k] +
             h2 * Wout[2 * 4 + k];
  }
}

extern "C" void kernel_launch(void* const* d_in, const int* in_sizes, int n_in,
                              void* d_out, int out_size, void* d_ws, size_t ws_size,
                              hipStream_t stream) {
  const float*     x    = (const float*)d_in[0];      // [N, 128]
  const long long* ei   = (const long long*)d_in[1];  // [2, E] int64
  const float*     W    = (const float*)d_in[2];      // [128, 3]
  const float*     b    = (const float*)d_in[3];      // [3]
  const float*     Wout = (const float*)d_in[4];      // [3, 4]
  const float*     bout = (const float*)d_in[5];      // [4]
  float* out = (float*)d_out;

  const int n_nodes = in_sizes[0] / IN_F;  // 100000
  const int n_edges = in_sizes[1] / 2;     // 3200000

  // Workspace layout (floats): xw[N*4] | deg[N] | dinv[N] | agg[N*4]  (~4 MB)
  float* ws   = (float*)d_ws;
  float* xw   = ws;
  float* deg  = xw + (size_t)n_nodes * 4;
  float* dinv = deg + n_nodes;
  float* agg  = dinv + n_nodes;

  const int tiles   = (n_nodes + 15) / 16;       // 16-row WMMA tiles
  int blocks1 = (tiles + 7) / 8;                 // 8 waves (tiles) per block
  const int blocksN = (n_nodes + 255) / 256;     // deg-init coverage
  if (blocks1 < blocksN) blocks1 = blocksN;
  const int blocksE = (n_edges + 255) / 256;

  gcn_xw_wmma <<<blocks1, 256, 0, stream>>>(x, W, xw, deg, n_nodes);
  gcn_degree  <<<blocksE, 256, 0, stream>>>(ei, deg, n_edges);
  gcn_selfloop<<<blocksN, 256, 0, stream>>>(deg, dinv, xw, agg, n_nodes);
  gcn_scatter <<<blocksE, 256, 0, stream>>>(ei, dinv, xw, agg, n_edges);
  gcn_out     <<<blocksN, 256, 0, stream>>>(agg, b, Wout, bout, out, n_nodes);
}